// _BaseLevel_23338852286540
// MI455X (gfx1250) — compile-verified
//
#include <hip/hip_runtime.h>

typedef float v4f __attribute__((ext_vector_type(4)));

// Instant-NGP hash primes (PRIMES[0] == 1, so x passes through unmultiplied)
#define PRIME_Y 2654435761u
#define PRIME_Z 805459861u

template <bool POW2>
__global__ __launch_bounds__(256) void hashgrid_trilerp_kernel(
    const float* __restrict__ pts,    // (N,3) f32
    const float* __restrict__ table,  // (buckets,8) f32
    float* __restrict__ out,          // (N,8) f32
    int N, unsigned buckets, unsigned mask)
{
    int n = blockIdx.x * blockDim.x + threadIdx.x;
    if (n >= N) return;

    // --- stream the point (read-once: non-temporal, keep L2 for the 128MB
    //     feature table which fully fits in the 192MB global L2) ---
    const float* p = pts + (size_t)n * 3;
    float px = __builtin_nontemporal_load(p + 0);
    float py = __builtin_nontemporal_load(p + 1);
    float pz = __builtin_nontemporal_load(p + 2);

    // continuous grid coords: exact f32 divide to bit-match the reference
    // (an ulp difference could flip floor() and pick a different hash slot)
    float qx = px / 0.005f;
    float qy = py / 0.005f;
    float qz = pz / 0.005f;

    float fbx = floorf(qx), fby = floorf(qy), fbz = floorf(qz);
    int bx = (int)fbx, by = (int)fby, bz = (int)fbz;
    float fx = qx - fbx, fy = qy - fby, fz = qz - fbz;

    // hash components for the two planes along each axis (uint32 wraparound)
    unsigned hx0 = (unsigned)bx;
    unsigned hx1 = (unsigned)(bx + 1);
    unsigned hy0 = (unsigned)by       * PRIME_Y;
    unsigned hy1 = (unsigned)(by + 1) * PRIME_Y;
    unsigned hz0 = (unsigned)bz       * PRIME_Z;
    unsigned hz1 = (unsigned)(bz + 1) * PRIME_Z;

    // corner c: bit0 -> x, bit1 -> y, bit2 -> z  (x-fastest, matches CORNERS)
    unsigned slot[8];
#pragma unroll
    for (int c = 0; c < 8; ++c) {
        unsigned h = ((c & 1) ? hx1 : hx0)
                   ^ ((c & 2) ? hy1 : hy0)
                   ^ ((c & 4) ? hz1 : hz0);
        slot[c] = POW2 ? (h & mask) : (h % buckets);
    }

    // --- issue all 16 b128 gathers up front: 16 outstanding LOADcnt entries
    //     per lane; rows are 32B-aligned; default RT temporal hint keeps the
    //     table L2-resident ---
    v4f lo[8], hi[8];
#pragma unroll
    for (int c = 0; c < 8; ++c) {
        const v4f* f = (const v4f*)(table + (size_t)slot[c] * 8);
        lo[c] = f[0];
        hi[c] = f[1];
    }

    // trilinear weights
    float wx[2] = {1.0f - fx, fx};
    float wy[2] = {1.0f - fy, fy};
    float wz[2] = {1.0f - fz, fz};

    // vector-form accumulate: lets the backend pack v_pk_fma_f32 / dual-FMAC
    v4f aclo = {0.0f, 0.0f, 0.0f, 0.0f};
    v4f achi = {0.0f, 0.0f, 0.0f, 0.0f};
#pragma unroll
    for (int c = 0; c < 8; ++c) {
        float w = wx[c & 1] * wy[(c >> 1) & 1] * wz[(c >> 2) & 1];
        aclo = lo[c] * w + aclo;
        achi = hi[c] * w + achi;
    }

    // streaming output: non-temporal b128 stores (write-once)
    v4f* o = (v4f*)(out + (size_t)n * 8);
    __builtin_nontemporal_store(aclo, o + 0);
    __builtin_nontemporal_store(achi, o + 1);
}

extern "C" void kernel_launch(void* const* d_in, const int* in_sizes, int n_in,
                              void* d_out, int out_size, void* d_ws, size_t ws_size,
                              hipStream_t stream) {
    const float* pts   = (const float*)d_in[0];   // (N,3) f32
    const float* table = (const float*)d_in[1];   // (buckets,8) f32
    float* out = (float*)d_out;                   // (N,8) f32

    int N = in_sizes[0] / 3;
    unsigned buckets = (unsigned)(in_sizes[1] / 8);
    bool pow2 = (buckets != 0u) && ((buckets & (buckets - 1u)) == 0u);
    unsigned mask = buckets - 1u;

    const int BLOCK = 256;
    int grid = (N + BLOCK - 1) / BLOCK;
    if (pow2) {
        hipLaunchKernelGGL(hashgrid_trilerp_kernel<true>, dim3(grid), dim3(BLOCK), 0, stream,
                           pts, table, out, N, buckets, mask);
    } else {
        hipLaunchKernelGGL(hashgrid_trilerp_kernel<false>, dim3(grid), dim3(BLOCK), 0, stream,
                           pts, table, out, N, buckets, mask);
    }
}